// RnnLayer_2894807957905
// MI455X (gfx1250) — compile-verified
//
#include <hip/hip_runtime.h>

typedef __attribute__((ext_vector_type(2))) float v2f;
typedef __attribute__((ext_vector_type(8))) float v8f;

// Problem dims (fixed by the reference)
static constexpr int T_STEPS = 512;
static constexpr int BATCH   = 128;
static constexpr int DIM     = 512;   // D_in == D_out == 512

// ---------------------------------------------------------------------------
// Register-blocked fp32 WMMA accumulation.
// C[j] += A[row0:row0+16, :] * Bmat[col0+16j : col0+16j+16, :]^T  (j = 0..NT-1)
// A is [M x 512] row-major; Bmat is [O x 512] row-major (we apply B^T), so
// effective B[k][n] for tile j = Bmat[(col0 + 16j + n)*512 + k].
//
// Per ISA layouts (wave32), V_WMMA_F32_16X16X4_F32:
//   A 16x4 f32 : lane L holds M = L&15 ; VGPR{0,1} hold K = 2*(L>>4)+{0,1}
//   B 4x16 f32 : lane L holds N = L&15 ; VGPR{0,1} hold K = 2*(L>>4)+{0,1}
//   C 16x16 f32: lane L holds N = L&15 ; VGPR v holds M = v + 8*(L>>4)
// One A load feeds NT WMMAs -> vmem:wmma = (NT+1):NT instead of 2:1.
// ---------------------------------------------------------------------------
template <int NT>
__device__ __forceinline__ void wmma_tile_accum_f32(const float* __restrict__ A,
                                                    int row0,
                                                    const float* __restrict__ Bmat,
                                                    int col0,
                                                    int lane, v8f (&c)[NT]) {
  const int mn = lane & 15;          // M for A-operand, N for B-operand
  const int kb = (lane >> 4) << 1;   // K sub-offset: 0 or 2
  const float* ap = A    + (size_t)(row0 + mn) * DIM + kb;
  const float* bp = Bmat + (size_t)(col0 + mn) * DIM + kb;
#pragma unroll 2
  for (int k0 = 0; k0 < DIM; k0 += 4) {
    v2f a = *(const v2f*)(ap + k0);
#pragma unroll
    for (int j = 0; j < NT; ++j) {
      v2f b = *(const v2f*)(bp + (size_t)(16 * j) * DIM + k0);
      c[j] = __builtin_amdgcn_wmma_f32_16x16x4_f32(
          /*neg_a=*/false, a, /*neg_b=*/false, b,
          /*c_mod=*/(short)0, c[j], /*reuse_a=*/false, /*reuse_b=*/false);
    }
  }
}

// ---------------------------------------------------------------------------
// Kernel 1: Y[(t*B+b), o] = sum_i X[(t*B+b), i] * W_in[o, i]  + b_h[o]
// 16x64 output strip per wave (4 N-tiles).  M tiles = 4096, N groups = 8.
// ---------------------------------------------------------------------------
__global__ void __launch_bounds__(128)
xproj_kernel(const float* __restrict__ X, const float* __restrict__ W_in,
             const float* __restrict__ b_h, float* __restrict__ Y) {
  const int lane = threadIdx.x & 31;
  const int wave = threadIdx.x >> 5;
  const int w    = blockIdx.x * 4 + wave;       // 32768 waves total
  const int ng   = w & 7;                       // 8 N-groups (64 cols each)
  const int mt   = w >> 3;                      // 4096 M tiles
  const int row0 = mt * 16;
  const int col0 = ng * 64;
  const int n    = lane & 15;
  const int hl   = lane >> 4;

  v8f c[4];
#pragma unroll
  for (int j = 0; j < 4; ++j) {
    const float bias = b_h[col0 + 16 * j + n];
#pragma unroll
    for (int v = 0; v < 8; ++v) c[j][v] = bias;
  }

  wmma_tile_accum_f32<4>(X, row0, W_in, col0, lane, c);

#pragma unroll
  for (int j = 0; j < 4; ++j) {
#pragma unroll
    for (int v = 0; v < 8; ++v) {
      const int row = row0 + v + 8 * hl;
      Y[(size_t)row * DIM + col0 + 16 * j + n] = c[j][v];
    }
  }
}

// ---------------------------------------------------------------------------
// Kernel 2: one recurrence step.
//   h_out = sigmoid( Xp_t + h_in @ W_h^T )           (pre-projected path)
//   h_out = sigmoid( b_h + X_t@W_in^T + h_in@W_h^T ) (fused fallback path)
// 16x32 strip per wave (2 N-tiles): 8 M-tiles x 16 N-groups = 128 waves,
// 32 blocks x 4 waves -- keeps parallelism high for the latency-bound scan.
// ---------------------------------------------------------------------------
__global__ void __launch_bounds__(128)
rnn_step_kernel(const float* __restrict__ h_in, float* __restrict__ h_out,
                const float* __restrict__ W_h,
                const float* __restrict__ Xp_t,   // proj+bias for this t (or null)
                const float* __restrict__ X_t,    // raw x_t (fused path)
                const float* __restrict__ W_in,
                const float* __restrict__ b_h,
                int fused) {
  const int lane = threadIdx.x & 31;
  const int wave = threadIdx.x >> 5;
  const int w    = blockIdx.x * 4 + wave;       // 128 waves
  const int ng   = w & 15;                      // 16 N-groups (32 cols each)
  const int mt   = w >> 4;                      // 8 M tiles
  const int row0 = mt * 16;
  const int col0 = ng * 32;
  const int n    = lane & 15;
  const int hl   = lane >> 4;

  v8f c[2];
  if (fused) {
#pragma unroll
    for (int j = 0; j < 2; ++j) {
      const float bias = b_h[col0 + 16 * j + n];
#pragma unroll
      for (int v = 0; v < 8; ++v) c[j][v] = bias;
    }
    wmma_tile_accum_f32<2>(X_t, row0, W_in, col0, lane, c);
  } else {
#pragma unroll
    for (int j = 0; j < 2; ++j)
#pragma unroll
      for (int v = 0; v < 8; ++v)
        c[j][v] = Xp_t[(size_t)(row0 + v + 8 * hl) * DIM + col0 + 16 * j + n];
  }

  wmma_tile_accum_f32<2>(h_in, row0, W_h, col0, lane, c);

#pragma unroll
  for (int j = 0; j < 2; ++j) {
#pragma unroll
    for (int v = 0; v < 8; ++v) {
      const int row = row0 + v + 8 * hl;
      const float x = c[j][v];
      h_out[(size_t)row * DIM + col0 + 16 * j + n] = 1.0f / (1.0f + __expf(-x));
    }
  }
}

// ---------------------------------------------------------------------------
// Host launcher.  Ping-pong h between d_out (even-step input, odd-step
// output) and d_ws; with 512 steps the final write (t=511, odd) lands in
// d_out.  Xproj lives in d_ws after the h buffer when it fits.
// ---------------------------------------------------------------------------
extern "C" void kernel_launch(void* const* d_in, const int* in_sizes, int n_in,
                              void* d_out, int out_size, void* d_ws, size_t ws_size,
                              hipStream_t stream) {
  const float* X    = (const float*)d_in[0];
  const float* W_in = (const float*)d_in[1];
  const float* W_h  = (const float*)d_in[2];
  const float* b_h  = (const float*)d_in[3];

  const size_t needH = (size_t)BATCH * DIM * sizeof(float);                 // 256 KB
  const size_t needY = (size_t)T_STEPS * BATCH * DIM * sizeof(float);       // 128 MB

  float* hA = (float*)d_out;                       // h at even steps; final output
  float* hB = (float*)d_ws;                        // h at odd steps
  float* Y  = (float*)((char*)d_ws + needH);
  const bool pre = ws_size >= needH + needY;

  // h0 = 0 (capture-legal memset node)
  hipMemsetAsync(d_out, 0, needH, stream);

  if (pre) {
    // 4096 M-tiles * 8 N-groups = 32768 waves, 4 waves/block
    xproj_kernel<<<32768 / 4, 128, 0, stream>>>(X, W_in, b_h, Y);
  }

  for (int t = 0; t < T_STEPS; ++t) {
    const float* h_in  = (t & 1) ? hB : hA;
    float*       h_out = (t & 1) ? hA : hB;
    const float* Xp_t  = pre ? (Y + (size_t)t * BATCH * DIM) : nullptr;
    const float* X_t   = X + (size_t)t * BATCH * DIM;
    rnn_step_kernel<<<32, 128, 0, stream>>>(h_in, h_out, W_h, Xp_t, X_t,
                                            W_in, b_h, pre ? 0 : 1);
  }
}